// SageGAT_2319282340413
// MI455X (gfx1250) — compile-verified
//
#include <hip/hip_runtime.h>
#include <hip/hip_bf16.h>

typedef float v2f __attribute__((ext_vector_type(2)));
typedef float v8f __attribute__((ext_vector_type(8)));

#define D 128
#define NCLS 16

static __device__ __forceinline__ void atomic_add_f32(float* p, float v) {
    __hip_atomic_fetch_add(p, v, __ATOMIC_RELAXED, __HIP_MEMORY_SCOPE_AGENT);
}

// ---- degree count: one thread per edge -------------------------------------
__global__ __launch_bounds__(256) void gnn_deg(const int* __restrict__ dst,
                                               float* __restrict__ deg, int E) {
    int i = blockIdx.x * 256 + threadIdx.x;
    if (i < E) atomic_add_f32(&deg[dst[i]], 1.0f);
}

// ---- edge scatter: 32 threads per edge, float4 chunk each ------------------
__global__ __launch_bounds__(256) void gnn_scatter(const float* __restrict__ feat,
                                                   const int* __restrict__ src,
                                                   const int* __restrict__ dst,
                                                   float* __restrict__ agg, int E) {
    int idx = blockIdx.x * 256 + threadIdx.x;
    int e = idx >> 5;
    if (e >= E) return;
    int c = (idx & 31) << 2;
    int s = src[e], d = dst[e];
    const float4 v = *(const float4*)(feat + (size_t)s * D + c);
    float* p = agg + (size_t)d * D + c;
    atomic_add_f32(p + 0, v.x);
    atomic_add_f32(p + 1, v.y);
    atomic_add_f32(p + 2, v.z);
    atomic_add_f32(p + 3, v.w);
}

// ---- layer 1: h = elu(mean_agg @ Wl + x @ Wr + b), fp32 WMMA ---------------
// one block = 16-node tile; 8 waves, wave w computes output columns [16w,16w+16)
__global__ __launch_bounds__(256) void gnn_layer1(
    const float* __restrict__ x, const float* __restrict__ agg,
    const float* __restrict__ deg, const float* __restrict__ Wl,
    const float* __restrict__ Wr, const float* __restrict__ bias,
    float* __restrict__ h, int Nn) {
    __shared__ float sA[16 * D];  // mean-scaled aggregate tile
    __shared__ float sX[16 * D];  // self-feature tile

    const int tid = threadIdx.x;
    const size_t base = (size_t)blockIdx.x * 16;

    // stage tiles into LDS: 512 float4 per matrix, 2 per thread
    for (int i = tid; i < 512; i += 256) {
        int r = i >> 5;   // row in tile
        int q = i & 31;   // float4 column
        size_t grow = base + r; if (grow >= (size_t)Nn) grow = (size_t)Nn - 1;
        size_t goff = grow * D + (size_t)q * 4;
        float inv = 1.0f / fmaxf(deg[grow], 1.0f);
        float4 a = *(const float4*)(agg + goff);
        a.x *= inv; a.y *= inv; a.z *= inv; a.w *= inv;
        ((float4*)sA)[i] = a;
        ((float4*)sX)[i] = *(const float4*)(x + goff);
    }
    __syncthreads();

    const int lane = tid & 31;
    const int wv   = tid >> 5;           // output column tile 0..7
    const int m    = lane & 15;          // row (A) / column (B,C)
    const int koff = (lane >> 4) * 2;    // K sub-offset per ISA A layout
    const int colg = wv * 16 + m;        // global output column

    v8f acc = {};
#pragma unroll
    for (int kk = 0; kk < 32; ++kk) {
        int k0 = kk * 4 + koff;
        v2f aA = *(const v2f*)(sA + m * D + k0);
        v2f aX = *(const v2f*)(sX + m * D + k0);
        v2f bL, bR;
        bL.x = Wl[(size_t)k0 * D + colg];
        bL.y = Wl[(size_t)(k0 + 1) * D + colg];
        bR.x = Wr[(size_t)k0 * D + colg];
        bR.y = Wr[(size_t)(k0 + 1) * D + colg];
        acc = __builtin_amdgcn_wmma_f32_16x16x4_f32(false, aA, false, bL,
                                                    (short)0, acc, false, false);
        acc = __builtin_amdgcn_wmma_f32_16x16x4_f32(false, aX, false, bR,
                                                    (short)0, acc, false, false);
    }

    const float bv = bias[colg];
    const int rowadd = (lane >> 4) * 8;  // C layout: VGPR v -> row v + 8*(lane/16)
#pragma unroll
    for (int vi = 0; vi < 8; ++vi) {
        size_t row = base + rowadd + vi;
        if (row < (size_t)Nn) {
            float v = acc[vi] + bv;
            v = (v > 0.0f) ? v : expm1f(v);   // ELU, alpha=1
            h[row * D + colg] = v;
        }
    }
}

// ---- layer 2: out = log_softmax(mean_agg2 @ Wl + h @ Wr + b) ---------------
// one wave per 16-node tile; single 16x16 output tile (NCLS == 16)
__global__ __launch_bounds__(32) void gnn_layer2(
    const float* __restrict__ h, const float* __restrict__ agg,
    const float* __restrict__ deg, const float* __restrict__ Wl,
    const float* __restrict__ Wr, const float* __restrict__ bias,
    float* __restrict__ out, int Nn) {
    const int lane = threadIdx.x;
    const size_t base = (size_t)blockIdx.x * 16;
    const int m    = lane & 15;
    const int koff = (lane >> 4) * 2;
    size_t grow = base + m; if (grow >= (size_t)Nn) grow = (size_t)Nn - 1;
    const size_t roff = grow * D;
    const float inv = 1.0f / fmaxf(deg[grow], 1.0f);

    v8f acc = {};
#pragma unroll
    for (int kk = 0; kk < 32; ++kk) {
        int k0 = kk * 4 + koff;
        v2f aA = *(const v2f*)(agg + roff + k0);
        aA.x *= inv; aA.y *= inv;
        v2f aH = *(const v2f*)(h + roff + k0);
        v2f bL, bR;
        bL.x = Wl[k0 * NCLS + m];
        bL.y = Wl[(k0 + 1) * NCLS + m];
        bR.x = Wr[k0 * NCLS + m];
        bR.y = Wr[(k0 + 1) * NCLS + m];
        acc = __builtin_amdgcn_wmma_f32_16x16x4_f32(false, aA, false, bL,
                                                    (short)0, acc, false, false);
        acc = __builtin_amdgcn_wmma_f32_16x16x4_f32(false, aH, false, bR,
                                                    (short)0, acc, false, false);
    }

    const float bv = bias[m];  // class = lane&15
    float v[8];
#pragma unroll
    for (int vi = 0; vi < 8; ++vi) v[vi] = acc[vi] + bv;

    // log_softmax over the 16 classes: classes live in the 16 lanes of a half
#pragma unroll
    for (int vi = 0; vi < 8; ++vi) {
        float rmax = v[vi];
        for (int s = 1; s < 16; s <<= 1) rmax = fmaxf(rmax, __shfl_xor(rmax, s, 32));
        float ssum = expf(v[vi] - rmax);
        for (int s = 1; s < 16; s <<= 1) ssum += __shfl_xor(ssum, s, 32);
        v[vi] = v[vi] - rmax - logf(ssum);
    }

    const int rowadd = (lane >> 4) * 8;
#pragma unroll
    for (int vi = 0; vi < 8; ++vi) {
        size_t row = base + rowadd + vi;
        if (row < (size_t)Nn) out[row * NCLS + m] = v[vi];
    }
}

extern "C" void kernel_launch(void* const* d_in, const int* in_sizes, int n_in,
                              void* d_out, int out_size, void* d_ws, size_t ws_size,
                              hipStream_t stream) {
    const float* x    = (const float*)d_in[0];
    const int*   ei   = (const int*)d_in[1];
    const float* W1_l = (const float*)d_in[2];
    const float* b1   = (const float*)d_in[3];
    const float* W1_r = (const float*)d_in[4];
    const float* W2_l = (const float*)d_in[5];
    const float* b2   = (const float*)d_in[6];
    const float* W2_r = (const float*)d_in[7];
    float* outp = (float*)d_out;

    const int N = in_sizes[0] / D;
    const int E = in_sizes[1] / 2;
    const int* src = ei;
    const int* dst = ei + E;

    // workspace layout (512-B aligned slabs)
    char* ws = (char*)d_ws;
    size_t off = 0;
    float* deg = (float*)(ws + off); off += (((size_t)N * 4 + 511) / 512) * 512;
    float* agg = (float*)(ws + off); off += (size_t)N * D * 4;
    float* hbuf = (float*)(ws + off);

    const int tiles = (N + 15) / 16;
    const int scat_blocks = (int)(((size_t)E * 32 + 255) / 256);

    // ---- layer 1 ----
    hipMemsetAsync(deg, 0, (size_t)N * 4, stream);
    hipMemsetAsync(agg, 0, (size_t)N * D * 4, stream);
    gnn_deg<<<(E + 255) / 256, 256, 0, stream>>>(dst, deg, E);
    gnn_scatter<<<scat_blocks, 256, 0, stream>>>(x, src, dst, agg, E);
    gnn_layer1<<<tiles, 256, 0, stream>>>(x, agg, deg, W1_l, W1_r, b1, hbuf, N);

    // ---- layer 2 ----
    hipMemsetAsync(agg, 0, (size_t)N * D * 4, stream);
    gnn_scatter<<<scat_blocks, 256, 0, stream>>>(hbuf, src, dst, agg, E);
    gnn_layer2<<<tiles, 32, 0, stream>>>(hbuf, agg, deg, W2_l, W2_r, b2, outp, N);
}